// TotalLoss_vis_atten_thm_fair_4_23991687315991
// MI455X (gfx1250) — compile-verified
//
#include <hip/hip_runtime.h>
#include <hip/hip_bf16.h>

typedef __attribute__((ext_vector_type(2))) float v2f;
typedef __attribute__((ext_vector_type(8))) float v8f;

#define NB       8
#define NOBJ     128
#define OUT_H    152
#define OUT_W    272
#define HW       41344            // 152*272
#define EMB      256
#define NID      537
#define NTILE_N  34               // ceil(537/16)
#define HM_BLKS  162              // ceil(HW/256)
#define EMB_SCALE 8.8871059f      // sqrt(2)*ln(536)
#define MIN_OVL  0.7f

struct WS {
  int   ind [NB][NOBJ];
  int   ids [NB][NOBJ];
  int   cxi [NB][NOBJ];
  int   cyi [NB][NOBJ];
  int   rvis[NB][NOBJ];
  float m   [NB][NOBJ];
  float v   [NB][NOBJ];
  float sg  [NB][NOBJ];
  float wh  [NB][NOBJ][8];
  float rg  [NB][NOBJ][4];
  float fpart[NB][HM_BLKS][3];    // pos_loss, neg_loss, npos per block
  float id_part[NB][8];           // per (sample, m-tile)
  float reg_l[NB], off_l[NB], vis_l[NB], cnt[NB];
};

__device__ __forceinline__ float gauss_radius(float height, float width) {
  float b1 = height + width;
  float c1 = width * height * (1.0f - MIN_OVL) / (1.0f + MIN_OVL);
  float r1 = (b1 + sqrtf(fmaxf(b1 * b1 - 4.0f * c1, 0.0f))) * 0.5f;
  float b2 = 2.0f * (height + width);
  float c2 = (1.0f - MIN_OVL) * width * height;
  float r2 = (b2 + sqrtf(fmaxf(b2 * b2 - 16.0f * c2, 0.0f))) * 0.5f;
  float b3 = -2.0f * MIN_OVL * (height + width);
  float c3 = (MIN_OVL - 1.0f) * width * height;
  float r3 = (b3 + sqrtf(fmaxf(b3 * b3 - 16.0f * MIN_OVL * c3, 0.0f))) * 0.5f;
  return fminf(fminf(r1, r2), r3);
}

// ---------------- 1: build per-object targets ----------------
__global__ __launch_bounds__(NOBJ)
void k_build(const float* __restrict__ ann1, const float* __restrict__ ann2, WS* S) {
  int b = blockIdx.x, i = threadIdx.x;
  const float* a1 = ann1 + ((size_t)b * NOBJ + i) * 7;
  float cx = a1[0] * (float)OUT_W, cy = a1[1] * (float)OUT_H;
  float w  = a1[2] * (float)OUT_W, h  = a1[3] * (float)OUT_H;
  float x1 = cx - w * 0.5f, y1 = cy - h * 0.5f;
  float x2 = x1 + w, y2 = y1 + h;
  float cxc = fminf(fmaxf(cx, 0.0f), (float)(OUT_W - 1));
  float cyc = fminf(fmaxf(cy, 0.0f), (float)(OUT_H - 1));
  float vis = a1[6];
  bool valid1 = (a1[4] != -1.0f);
  float r0 = gauss_radius(ceilf(h), ceilf(w));
  int r_vis = (int)floorf(fmaxf(r0 * expf(vis - 1.0f), 0.0f));
  int cxi = (int)cxc, cyi = (int)cyc;

  // first matching id in ann2 (valid rows only)
  float id1 = a1[5];
  const float* a2b = ann2 + (size_t)b * NOBJ * 7;
  int midx = 0; bool has_match = false;
  for (int j = 0; j < NOBJ; ++j) {
    const float* r = a2b + (size_t)j * 7;
    if (!has_match && r[5] == id1 && r[4] != -1.0f) { midx = j; has_match = true; }
  }
  const float* a2 = a2b + (size_t)midx * 7;
  float cx2 = a2[0] * (float)OUT_W, cy2 = a2[1] * (float)OUT_H;
  float w2  = a2[2] * (float)OUT_W, h2  = a2[3] * (float)OUT_H;
  float x12 = cx2 - w2 * 0.5f, y12 = cy2 - h2 * 0.5f;
  float x22 = x12 + w2, y22 = y12 + h2;
  float cxc2 = fminf(fmaxf(cx2, 0.0f), (float)(OUT_W - 1));
  float cyc2 = fminf(fmaxf(cy2, 0.0f), (float)(OUT_H - 1));

  bool ok = valid1 && has_match && (h > 0.0f) && (w > 0.0f) && (h2 > 0.0f) && (w2 > 0.0f);
  float m = ok ? 1.0f : 0.0f;

  S->ind [b][i] = cyi * OUT_W + cxi;
  S->ids [b][i] = ok ? (int)id1 : 0;
  S->cxi [b][i] = cxi;  S->cyi[b][i] = cyi;  S->rvis[b][i] = r_vis;
  S->m   [b][i] = m;
  S->v   [b][i] = vis * m;
  S->sg  [b][i] = (2.0f * (float)r_vis + 1.0f) / 6.0f;
  float whv[8] = { cxc - x1, cyc - y1, x2 - cxc, y2 - cyc,
                   cxc - x12, cyc - y12, x22 - cxc, y22 - cyc };
  float rgv[4] = { cxc - (float)cxi, cyc - (float)cyi,
                   cxc2 - (float)cxi, cyc2 - (float)cyi };
  for (int c = 0; c < 8; ++c) S->wh[b][i][c] = whv[c] * m;
  for (int c = 0; c < 4; ++c) S->rg[b][i][c] = rgv[c] * m;
}

// ---------------- 2: heatmap + focal partials ----------------
__global__ __launch_bounds__(256)
void k_hm(const float* __restrict__ cls, WS* S) {
  __shared__ int   scx[NOBJ], scy[NOBJ], srr[NOBJ];
  __shared__ float smm[NOBJ], ssg[NOBJ];
  __shared__ float rA[256], rB[256], rC[256];
  int b = blockIdx.y;
  int tid = threadIdx.x;
  if (tid < NOBJ) {
    scx[tid] = S->cxi[b][tid]; scy[tid] = S->cyi[b][tid];
    srr[tid] = S->rvis[b][tid];
    smm[tid] = S->m[b][tid];   ssg[tid] = S->sg[b][tid];
  }
  __syncthreads();
  int p = blockIdx.x * 256 + tid;
  float pos_l = 0.0f, neg_l = 0.0f, npos = 0.0f;
  if (p < HW) {
    int y = p / OUT_W, x = p - y * OUT_W;
    float hm = 0.0f;
    for (int i = 0; i < NOBJ; ++i) {
      if (smm[i] == 0.0f) continue;
      float dx = (float)(x - scx[i]);
      float dy = (float)(y - scy[i]);
      float rf = (float)srr[i];
      if (fabsf(dx) <= rf && fabsf(dy) <= rf) {
        float s = ssg[i];
        float g = expf(-(dx * dx + dy * dy) / (2.0f * s * s));
        hm = fmaxf(hm, g);
      }
    }
    float c = cls[(size_t)b * HW + p];
    float pr = 1.0f / (1.0f + expf(-c));
    pr = fminf(fmaxf(pr, 0.0001f), 0.9999f);
    if (hm == 1.0f) {
      float om = 1.0f - pr;
      pos_l = logf(pr) * om * om;
      npos = 1.0f;
    } else {
      float og = 1.0f - hm; float og2 = og * og;
      neg_l = logf(1.0f - pr) * pr * pr * og2 * og2;
    }
  }
  rA[tid] = pos_l; rB[tid] = neg_l; rC[tid] = npos;
  __syncthreads();
  for (int s = 128; s > 0; s >>= 1) {
    if (tid < s) { rA[tid] += rA[tid + s]; rB[tid] += rB[tid + s]; rC[tid] += rC[tid + s]; }
    __syncthreads();
  }
  if (tid == 0) {
    S->fpart[b][blockIdx.x][0] = rA[0];
    S->fpart[b][blockIdx.x][1] = rB[0];
    S->fpart[b][blockIdx.x][2] = rC[0];
  }
}

// ---------------- 3: gathered L1 losses ----------------
__global__ __launch_bounds__(NOBJ)
void k_reg(const float* __restrict__ regs, const float* __restrict__ offs,
           const float* __restrict__ viss, WS* S) {
  __shared__ float r0[NOBJ], r1[NOBJ], r2[NOBJ], r3[NOBJ];
  int b = blockIdx.x, i = threadIdx.x;
  int ind = S->ind[b][i];
  float m = S->m[b][i];
  float sreg = 0.0f;
  for (int c = 0; c < 8; ++c) {
    float pv = regs[((size_t)b * 8 + c) * HW + ind] * m;
    sreg += fabsf(pv - S->wh[b][i][c] * m);
  }
  float soff = 0.0f;
  for (int c = 0; c < 4; ++c) {
    float pv = offs[((size_t)b * 4 + c) * HW + ind] * m;
    soff += fabsf(pv - S->rg[b][i][c] * m);
  }
  float vp = 1.0f / (1.0f + expf(-viss[(size_t)b * HW + ind]));
  vp = fminf(fmaxf(vp, 0.0001f), 0.9999f);
  float svis = fabsf(vp * m - S->v[b][i] * m);
  r0[i] = sreg; r1[i] = soff; r2[i] = svis; r3[i] = m;
  __syncthreads();
  for (int s = 64; s > 0; s >>= 1) {
    if (i < s) { r0[i] += r0[i + s]; r1[i] += r1[i + s]; r2[i] += r2[i + s]; r3[i] += r3[i + s]; }
    __syncthreads();
  }
  if (i == 0) {
    float cnt = r3[0];
    S->reg_l[b] = r0[0] / (cnt * 8.0f + 0.0001f);
    S->off_l[b] = r1[0] / (cnt * 4.0f + 0.0001f);
    S->vis_l[b] = r2[0] / (cnt * 1.0f + 0.0001f);
    S->cnt[b]   = cnt;
  }
}

// ---------------- 4: ID loss via WMMA f32 16x16x4 ----------------
// block = (sample b, M-tile mt): 16 rows of 128x537 logits, 8 waves sweep 34 N-tiles.
// Pad columns (537..543) use a clamped Wc row; their logits land in sL slots that
// the log-softmax never reads, so no EXEC divergence is needed in the k-loop.
__global__ __launch_bounds__(256)
void k_id(const float* __restrict__ identis, const float* __restrict__ Wc,
          const float* __restrict__ bc, WS* S) {
  __shared__ float sA[16][258];      // normalized embeddings (padded, 8B-aligned rows)
  __shared__ float sL[16][545];      // logits 16 x 544 (padded)
  __shared__ float snrm[16];
  __shared__ int   sind[16];
  __shared__ float spart[16];
  int b  = blockIdx.x >> 3;
  int mt = blockIdx.x & 7;
  int tid = threadIdx.x;

  if (tid < 16) sind[tid] = S->ind[b][mt * 16 + tid];
  __syncthreads();
  for (int idx = tid; idx < 16 * EMB; idx += 256) {
    int mm = idx >> 8, c = idx & 255;
    sA[mm][c] = identis[(size_t)b * EMB * HW + (size_t)c * HW + sind[mm]];
  }
  __syncthreads();
  if (tid < 16) {
    float s = 0.0f;
    for (int c = 0; c < EMB; ++c) s += sA[tid][c] * sA[tid][c];
    snrm[tid] = sqrtf(s);
  }
  __syncthreads();
  for (int idx = tid; idx < 16 * EMB; idx += 256) {
    int mm = idx >> 8, c = idx & 255;
    sA[mm][c] = sA[mm][c] * (EMB_SCALE / fmaxf(snrm[mm], 1e-12f));
  }
  __syncthreads();

  int wv = tid >> 5, lane = tid & 31;
  int li = lane & 15;                 // M index for A, N index for B/D
  int koff = (lane >> 4) << 1;        // K offset {0,2} per half-wave
  const float* aRow = &sA[li][koff];  // 8B-aligned (row stride 258, koff even)

  for (int nt = wv; nt < NTILE_N; nt += 8) {
    int colb = nt * 16;
    int col = colb + li;
    int colc = (col < NID) ? col : (NID - 1);   // clamp: pad cols read row 536
    const float* wrow = Wc + (size_t)colc * EMB + koff;  // 8B-aligned

    v8f acc = {0.f, 0.f, 0.f, 0.f, 0.f, 0.f, 0.f, 0.f};
    // software-pipelined: fetch k+4 operands while WMMA consumes k
    v2f av = *(const v2f*)(aRow);
    v2f bv = *(const v2f*)(wrow);
#pragma unroll 4
    for (int k = 0; k < EMB - 4; k += 4) {
      v2f av2 = *(const v2f*)(aRow + k + 4);
      v2f bv2 = *(const v2f*)(wrow + k + 4);
      acc = __builtin_amdgcn_wmma_f32_16x16x4_f32(
          false, av, false, bv, (short)0, acc, false, false);
      av = av2; bv = bv2;
    }
    acc = __builtin_amdgcn_wmma_f32_16x16x4_f32(
        false, av, false, bv, (short)0, acc, false, false);

    float bias = bc[colc];             // pad cols get bc[536]; never read back
    int rbase = (lane >> 4) * 8;       // D: lanes 16-31 hold M = r+8
#pragma unroll
    for (int r = 0; r < 8; ++r) sL[rbase + r][col] = acc[r] + bias;
  }
  __syncthreads();

  if (tid < 16) {
    int grow = mt * 16 + tid;
    float mx = -3.4e38f;
    for (int n = 0; n < NID; ++n) mx = fmaxf(mx, sL[tid][n]);
    float sum = 0.0f;
    for (int n = 0; n < NID; ++n) sum += expf(sL[tid][n] - mx);
    float lse = mx + logf(sum);
    int gid = S->ids[b][grow];
    float picked = sL[tid][gid] - lse;
    spart[tid] = S->m[b][grow] * S->v[b][grow] * picked;
  }
  __syncthreads();
  if (tid == 0) {
    float s = 0.0f;
    for (int r = 0; r < 16; ++r) s += spart[r];
    S->id_part[b][mt] = s;
  }
}

// ---------------- 5: finalize ----------------
__global__ __launch_bounds__(32)
void k_fin(const WS* S, const float* __restrict__ sdet, const float* __restrict__ sid,
           float* __restrict__ out) {
  __shared__ float cl[NB], il[NB];
  int t = threadIdx.x;
  if (t < NB) {
    float pos = 0.0f, neg = 0.0f, np = 0.0f;
    for (int k = 0; k < HM_BLKS; ++k) {
      pos += S->fpart[t][k][0];
      neg += S->fpart[t][k][1];
      np  += S->fpart[t][k][2];
    }
    cl[t] = (np > 0.0f) ? (-(pos + neg) / fmaxf(np, 1.0f)) : (-neg);
    float num = 0.0f;
    for (int k = 0; k < 8; ++k) num += S->id_part[t][k];
    float cnt = S->cnt[t];
    il[t] = (cnt > 0.0f) ? (-num / fmaxf(cnt, 1.0f)) : 0.0f;
  }
  __syncthreads();
  if (t == 0) {
    float sc = 0, sr = 0, so = 0, sv = 0, si = 0;
    for (int bb = 0; bb < NB; ++bb) {
      sc += cl[bb]; sr += S->reg_l[bb]; so += S->off_l[bb];
      sv += S->vis_l[bb]; si += il[bb];
    }
    out[0] = sc / (float)NB;
    out[1] = sr / (float)NB;
    out[2] = so / (float)NB;
    out[3] = sv / (float)NB;
    out[4] = si / (float)NB;
    out[5] = sdet[0];
    out[6] = sid[0];
  }
}

extern "C" void kernel_launch(void* const* d_in, const int* in_sizes, int n_in,
                              void* d_out, int out_size, void* d_ws, size_t ws_size,
                              hipStream_t stream) {
  const float* cls   = (const float*)d_in[0];   // [8,1,HW]
  const float* regs  = (const float*)d_in[1];   // [8,8,HW]
  const float* offs  = (const float*)d_in[2];   // [8,4,HW]
  const float* viss  = (const float*)d_in[3];   // [8,1,HW]
  const float* ident = (const float*)d_in[4];   // [8,256,HW]
  const float* ann1  = (const float*)d_in[5];   // [8,128,7]
  const float* ann2  = (const float*)d_in[6];   // [8,128,7]
  const float* sdet  = (const float*)d_in[7];   // [1]
  const float* sid   = (const float*)d_in[8];   // [1]
  const float* Wc    = (const float*)d_in[9];   // [537,256]
  const float* bc    = (const float*)d_in[10];  // [537]
  float* out = (float*)d_out;                   // 7 scalars
  WS* S = (WS*)d_ws;

  k_build<<<NB, NOBJ, 0, stream>>>(ann1, ann2, S);
  k_hm<<<dim3(HM_BLKS, NB), 256, 0, stream>>>(cls, S);
  k_reg<<<NB, NOBJ, 0, stream>>>(regs, offs, viss, S);
  k_id<<<NB * 8, 256, 0, stream>>>(ident, Wc, bc, S);
  k_fin<<<1, 32, 0, stream>>>(S, sdet, sid, out);
}